// SSIMLoss_67929202754041
// MI455X (gfx1250) — compile-verified
//
#include <hip/hip_runtime.h>

typedef float v2f __attribute__((ext_vector_type(2)));
typedef float v8f __attribute__((ext_vector_type(8)));

#define WIN 7
#define IMG 384
#define OUTD 378
#define TILES_SIDE 24
#define TILES_IMG (TILES_SIDE * TILES_SIDE)
#define N_IMG 64
#define TOTAL_TILES (N_IMG * TILES_IMG)   // 36864
#define WAVES_PER_BLOCK 8
#define LDS_ROWS 22
#define LDS_STRIDE 24                      // 22 cols padded -> fewer bank conflicts

__global__ void ssim_zero_ws(float* ws) { ws[0] = 0.0f; }

__global__ __launch_bounds__(256)
void ssim_main(const float* __restrict__ x, const float* __restrict__ y,
               const float* __restrict__ dr, float* __restrict__ ws) {
  __shared__ float lx[WAVES_PER_BLOCK][LDS_ROWS * LDS_STRIDE];
  __shared__ float ly[WAVES_PER_BLOCK][LDS_ROWS * LDS_STRIDE];
  __shared__ float red[WAVES_PER_BLOCK];

  const int tid  = threadIdx.x;
  const int wave = tid >> 5;
  const int lane = tid & 31;

  const int tile = blockIdx.x * WAVES_PER_BLOCK + wave;   // grid sized exactly
  const int img  = tile / TILES_IMG;
  const int t    = tile % TILES_IMG;
  const int r0   = (t / TILES_SIDE) * 16;                 // output-tile origin
  const int c0   = (t % TILES_SIDE) * 16;

  const float* xb = x + (size_t)img * IMG * IMG;
  const float* yb = y + (size_t)img * IMG * IMG;

  // Stage the 22x22 halo tiles of x and y straight into LDS with the CDNA5
  // async global->LDS path (ASYNCcnt). Per-lane b32 transfers, edge-clamped
  // per element so every address stays inside the allocation; clamped
  // elements only feed outputs that are masked out below.
  for (int i = lane; i < LDS_ROWS * 22; i += 32) {
    int row = i / 22, col = i % 22;
    int gr = r0 + row; if (gr > IMG - 1) gr = IMG - 1;
    int gc = c0 + col; if (gc > IMG - 1) gc = IMG - 1;
    size_t g = (size_t)gr * IMG + gc;
    const float* gx = xb + g;
    const float* gy = yb + g;
    unsigned dx = (unsigned)(uintptr_t)&lx[wave][row * LDS_STRIDE + col];
    unsigned dy = (unsigned)(uintptr_t)&ly[wave][row * LDS_STRIDE + col];
    asm volatile("global_load_async_to_lds_b32 %0, %1, off"
                 :: "v"(dx), "v"(gx) : "memory");
    asm volatile("global_load_async_to_lds_b32 %0, %1, off"
                 :: "v"(dy), "v"(gy) : "memory");
  }
  // Wait for this wave's async transfers; each wave consumes only its slice.
  asm volatile("s_wait_asynccnt 0" ::: "memory");
  __syncthreads();

  const int myc = lane & 15;   // output column within tile / B,C,D lane column
  const int p   = lane >> 4;   // lane-half
  const int m   = lane & 15;   // A-matrix row owned by this lane

  v8f Sx = {}, Sy = {}, Sxx = {}, Syy = {}, Sxy = {};

  // Vertical 7-tap sum as D += M(16x24 band) * H(24x16), six K=4 chunks on
  // the f32 matrix pipe (exact arithmetic, no precision loss vs reference).
#pragma unroll
  for (int k = 0; k < 6; ++k) {
    // A fragment: A[m][kk] = 1 iff m <= kk < m+7.
    // Layout: lanes 0-15 hold {K=4k,4k+1}, lanes 16-31 hold {K=4k+2,4k+3}.
    v2f Ak;
#pragma unroll
    for (int e = 0; e < 2; ++e) {
      int kk = 4 * k + 2 * p + e;
      Ak[e] = (kk >= m && kk < m + WIN) ? 1.0f : 0.0f;
    }
    // B fragments: horizontal 7-sums of x, y, xx, yy, xy.
    // Layout: V0 -> rows {4k (lanes0-15), 4k+1 (lanes16-31)}, V1 -> {4k+2, 4k+3}.
    v2f Bx, By, Bxx, Byy, Bxy;
#pragma unroll
    for (int v = 0; v < 2; ++v) {
      int row = 4 * k + 2 * v + p;         // H-row this lane supplies
      float hx = 0.f, hy = 0.f, hxx = 0.f, hyy = 0.f, hxy = 0.f;
      if (row < LDS_ROWS) {
        const float* rx = &lx[wave][row * LDS_STRIDE + myc];
        const float* ry = &ly[wave][row * LDS_STRIDE + myc];
#pragma unroll
        for (int j = 0; j < WIN; ++j) {
          float xv = rx[j], yv = ry[j];
          hx += xv; hy += yv;
          hxx = __builtin_fmaf(xv, xv, hxx);
          hyy = __builtin_fmaf(yv, yv, hyy);
          hxy = __builtin_fmaf(xv, yv, hxy);
        }
      }
      Bx[v] = hx; By[v] = hy; Bxx[v] = hxx; Byy[v] = hyy; Bxy[v] = hxy;
    }
    Sx  = __builtin_amdgcn_wmma_f32_16x16x4_f32(false, Ak, false, Bx,  (short)0, Sx,  false, false);
    Sy  = __builtin_amdgcn_wmma_f32_16x16x4_f32(false, Ak, false, By,  (short)0, Sy,  false, false);
    Sxx = __builtin_amdgcn_wmma_f32_16x16x4_f32(false, Ak, false, Bxx, (short)0, Sxx, false, false);
    Syy = __builtin_amdgcn_wmma_f32_16x16x4_f32(false, Ak, false, Byy, (short)0, Syy, false, false);
    Sxy = __builtin_amdgcn_wmma_f32_16x16x4_f32(false, Ak, false, Bxy, (short)0, Sxy, false, false);
  }

  const float inv49 = 1.0f / 49.0f;
  const float covn  = 49.0f / 48.0f;
  const float drv   = dr[img];
  const float c1    = (0.01f * drv) * (0.01f * drv);
  const float c2    = (0.03f * drv) * (0.03f * drv);

  // C/D layout: VGPR g -> output row g + 8*p, column = myc.
  float lsum = 0.0f;
#pragma unroll
  for (int g = 0; g < 8; ++g) {
    int orow = r0 + g + 8 * p;
    int ocol = c0 + myc;
    float ux  = Sx[g]  * inv49, uy  = Sy[g]  * inv49;
    float uxx = Sxx[g] * inv49, uyy = Syy[g] * inv49, uxy = Sxy[g] * inv49;
    float vx  = covn * (uxx - ux * ux);
    float vy  = covn * (uyy - uy * uy);
    float vxy = covn * (uxy - ux * uy);
    float a1 = 2.0f * ux * uy + c1;
    float a2 = 2.0f * vxy + c2;
    float b1 = ux * ux + uy * uy + c1;
    float b2 = vx + vy + c2;
    if (orow < OUTD && ocol < OUTD) lsum += (a1 * a2) / (b1 * b2);
  }

  // wave32 reduction, then block reduction, one atomic per block.
#pragma unroll
  for (int o = 16; o > 0; o >>= 1) lsum += __shfl_xor(lsum, o, 32);
  if (lane == 0) red[wave] = lsum;
  __syncthreads();
  if (tid == 0) {
    float bsum = 0.0f;
#pragma unroll
    for (int w = 0; w < WAVES_PER_BLOCK; ++w) bsum += red[w];
    atomicAdd(ws, bsum);
  }
}

__global__ void ssim_final(const float* __restrict__ ws, float* __restrict__ out) {
  out[0] = 1.0f - ws[0] * (1.0f / (float)(N_IMG * OUTD * OUTD));
}

extern "C" void kernel_launch(void* const* d_in, const int* in_sizes, int n_in,
                              void* d_out, int out_size, void* d_ws, size_t ws_size,
                              hipStream_t stream) {
  const float* x  = (const float*)d_in[0];
  const float* y  = (const float*)d_in[1];
  const float* dr = (const float*)d_in[2];
  float* ws  = (float*)d_ws;
  float* out = (float*)d_out;

  ssim_zero_ws<<<1, 1, 0, stream>>>(ws);
  ssim_main<<<TOTAL_TILES / WAVES_PER_BLOCK, 256, 0, stream>>>(x, y, dr, ws);
  ssim_final<<<1, 1, 0, stream>>>(ws, out);
}